// VAEChamferDistance_51015621542387
// MI455X (gfx1250) — compile-verified
//
#include <hip/hip_runtime.h>
#include <hip/hip_bf16.h>
#include <stdint.h>

#define B_      32
#define N_      2048
#define D_      3
#define LATENT_ 256
#define NTILES  (N_ / 16)          // 128 ref tiles of 16 points
#define WPB     4                  // waves per block (4 x 16 queries)
#define TPB     (WPB * 32)

typedef float v2f __attribute__((ext_vector_type(2)));
typedef float v8f __attribute__((ext_vector_type(8)));
typedef unsigned int u32x4 __attribute__((ext_vector_type(4)));
typedef int i32x4 __attribute__((ext_vector_type(4)));
typedef int i32x8 __attribute__((ext_vector_type(8)));

#if defined(__has_builtin)
#if __has_builtin(__builtin_amdgcn_tensor_load_to_lds) && \
    __has_builtin(__builtin_amdgcn_s_wait_tensorcnt)
#define HAVE_TDM 1
#endif
#endif

__global__ void zero_out_kernel(float* out) {
    if (threadIdx.x == 0) out[0] = 0.0f;
}

// One wave owns a 16-query tile. Wave 0 DMAs the full reference cloud for
// (batch, direction) into LDS via the Tensor Data Mover; the block then
// rewrites it once into WMMA-B layout; each wave sweeps 128 ref tiles with
// V_WMMA_F32_16X16X4_F32 computing  -2*q.p + |p|^2  per 16x16 tile.
__global__ void __launch_bounds__(TPB)
chamfer_kernel(const float* __restrict__ in_pts,
               const float* __restrict__ out_pts,
               float* __restrict__ out_scalar) {
    __shared__ float refsRaw[N_ * D_];     // 24 KB: raw (x,y,z) x 2048
    __shared__ v2f   refsB[NTILES * 32];   // 32 KB: B-operand layout per tile

    const int batch = blockIdx.y;          // 0..30 (reference skips batch 31)
    const int dir   = blockIdx.z;          // 0: q=in refs=out ; 1: swapped
    const float* Qb = (dir == 0 ? in_pts  : out_pts) + (size_t)batch * N_ * D_;
    const float* Pb = (dir == 0 ? out_pts : in_pts ) + (size_t)batch * N_ * D_;

    const int tid  = threadIdx.x;          // 0..127
    const int wave = tid >> 5;
    const int lane = tid & 31;

#ifdef HAVE_TDM
    // --- Tensor Data Mover: 1-tile DMA of 6144 f32 (24 KB) global -> LDS ---
    if (wave == 0) {
        const uint64_t ga = (uint64_t)(uintptr_t)Pb;
        const uint32_t la = (uint32_t)(uintptr_t)refsRaw;   // flat low-32 = LDS offset
        u32x4 g0;
        g0[0] = 1u;                                    // count=1, user descriptor
        g0[1] = la;                                    // lds_addr [63:32]
        g0[2] = (uint32_t)ga;                          // global_addr [95:64]
        g0[3] = ((uint32_t)(ga >> 32) & 0x01FFFFFFu)   // global_addr [120:96]
              | 0x80000000u;                           // type=2 ("image") [127:126]
        i32x8 g1;
        g1[0] = (int)(2u << 16);                       // data_size=2 (4B); mask=0
        g1[1] = (int)((unsigned)(N_ * D_) << 16);      // tensor_dim0[15:0] @ [63:48]
        g1[2] = (int)(((N_ * D_) >> 16) & 0xFFFF)      // tensor_dim0[31:16]
              | (1 << 16);                             // tensor_dim1 = 1
        g1[3] = (int)((unsigned)(N_ * D_) << 16);      // tile_dim0 = 6144 @ [127:112]
        g1[4] = 1;                                     // tile_dim1 = 1
        g1[5] = N_ * D_;                               // tensor_dim0_stride
        g1[6] = 0;
        g1[7] = 0;
        const i32x4 gz = {0, 0, 0, 0};                 // <=2D: groups 2/3 unused
#if __clang_major__ >= 23
        const i32x8 gz8 = {0, 0, 0, 0, 0, 0, 0, 0};
        __builtin_amdgcn_tensor_load_to_lds(g0, g1, gz, gz, gz8, 0);
#else
        __builtin_amdgcn_tensor_load_to_lds(g0, g1, gz, gz, 0);
#endif
        __builtin_amdgcn_s_wait_tensorcnt((short)0);
    }
    __syncthreads();
#else
    for (int i = tid; i < N_ * D_; i += TPB) refsRaw[i] = Pb[i];
    __syncthreads();
#endif

    // --- Build WMMA B-operand layout once per block (shared by all 4 waves):
    // slot (tile, l): l<16 -> (px,py) of point tile*16+l ; l>=16 -> (pz,|p|^2).
    for (int idx = tid; idx < NTILES * 32; idx += TPB) {
        const int sl = idx & 31;
        const int p  = (idx >> 5) * 16 + (sl & 15);
        const float px = refsRaw[3 * p + 0];
        const float py = refsRaw[3 * p + 1];
        const float pz = refsRaw[3 * p + 2];
        const float psq = px * px + py * py + pz * pz;
        v2f bb;
        bb.x = (sl >= 16) ? pz  : px;
        bb.y = (sl >= 16) ? psq : py;
        refsB[idx] = bb;
    }
    __syncthreads();

    // --- Query tile: A row (M = lane%16) = (-2qx,-2qy,-2qz,1);
    // lanes 0-15 carry (K0,K1), lanes 16-31 carry (K2,K3).
    const bool hi   = lane >= 16;
    const int  qidx = (blockIdx.x * WPB + wave) * 16 + (lane & 15);
    const float qx = Qb[3 * qidx + 0];
    const float qy = Qb[3 * qidx + 1];
    const float qz = Qb[3 * qidx + 2];
    const float qsq = qx * qx + qy * qy + qz * qz;
    v2f a;
    a.x = hi ? (-2.0f * qz) : (-2.0f * qx);
    a.y = hi ? 1.0f         : (-2.0f * qy);

    v8f runmin;
    #pragma unroll
    for (int r = 0; r < 8; ++r) runmin[r] = 3.4e38f;
    const v8f czero = {};

    // --- Steady state: ds_load_b64 + v_wmma + elementwise min, 128 tiles ---
    for (int t = 0; t < NTILES; ++t) {
        const v2f b = refsB[t * 32 + lane];
        v8f d = __builtin_amdgcn_wmma_f32_16x16x4_f32(
            /*neg_a=*/false, a, /*neg_b=*/false, b,
            /*c_mod=*/(short)0, czero, /*reuse_a=*/false, /*reuse_b=*/false);
        #pragma unroll
        for (int r = 0; r < 8; ++r) runmin[r] = fminf(runmin[r], d[r]);
    }

    // --- Per-row min over N (C/D layout: VGPR r, lane l -> M=r+8*(l>=16),
    // N=l%16): butterfly min within each 16-lane half, then sum rows.
    float acc = 0.0f;
    #pragma unroll
    for (int r = 0; r < 8; ++r) {
        float v = runmin[r];
        #pragma unroll
        for (int m = 1; m < 16; m <<= 1) v = fminf(v, __shfl_xor(v, m, 32));
        acc += v;                          // lower half: rows 0-7; upper: 8-15
    }
    acc += __shfl_xor(acc, 16, 32);        // all lanes: sum of 16 row-mins

    // Add |q|^2 per row: halves duplicate queries -> 0.5 * full-wave sum.
    float qs = qsq;
    #pragma unroll
    for (int m = 1; m < 32; m <<= 1) qs += __shfl_xor(qs, m, 32);

    const float total = acc + 0.5f * qs;   // sum_M [ min_N d + |q_M|^2 ]
    if (lane == 0) atomicAdd(out_scalar, total);
}

// out += ALFA * 0.5 * sum(exp(v) + m^2 - v)  == 0.25 * sum(...)
__global__ void kl_kernel(const float* __restrict__ mean,
                          const float* __restrict__ var,
                          float* __restrict__ out_scalar) {
    const int i = blockIdx.x * blockDim.x + threadIdx.x;
    float t = 0.0f;
    if (i < B_ * LATENT_) {
        const float m = mean[i];
        const float v = var[i];
        t = __expf(v) + m * m - v;
    }
    #pragma unroll
    for (int m = 1; m < 32; m <<= 1) t += __shfl_xor(t, m, 32);
    if ((threadIdx.x & 31) == 0) atomicAdd(out_scalar, 0.25f * t);
}

extern "C" void kernel_launch(void* const* d_in, const int* in_sizes, int n_in,
                              void* d_out, int out_size, void* d_ws, size_t ws_size,
                              hipStream_t stream) {
    const float* in_points  = (const float*)d_in[0];
    // d_in[1] = in_batch (int64, unused: equal-sized sorted batches)
    const float* out_points = (const float*)d_in[2];
    // d_in[3] = out_batch (unused)
    const float* mean       = (const float*)d_in[4];
    const float* variance   = (const float*)d_in[5];
    float* out = (float*)d_out;

    zero_out_kernel<<<1, 32, 0, stream>>>(out);

    dim3 grid(N_ / (16 * WPB), B_ - 1, 2);   // (32, 31, 2)
    chamfer_kernel<<<grid, TPB, 0, stream>>>(in_points, out_points, out);

    kl_kernel<<<(B_ * LATENT_ + 255) / 256, 256, 0, stream>>>(mean, variance, out);
}